// CompetitiveLayer_45518063403577
// MI455X (gfx1250) — compile-verified
//
#include <hip/hip_runtime.h>

typedef __attribute__((ext_vector_type(2))) float v2f;
typedef __attribute__((ext_vector_type(8))) float v8f;

#define NA 4096
#define NB 4096
#define N_ITERS 64

// ---------------- init: zero accumulators/counters, AF = AT ----------------
__global__ void cl_init(const float* __restrict__ AT,
                        float* __restrict__ S, float* __restrict__ T,
                        float* __restrict__ AF, int* __restrict__ counters) {
  int i = blockIdx.x * blockDim.x + threadIdx.x;
  if (i < NA) {
    S[i] = 0.0f;
    T[i] = 0.0f;
    AF[i] = AT[i];
  }
  if (i == 0) { counters[0] = 0; counters[1] = 0; }
}

// ---------------- colsum: S[j] += sum_i k[i][j]^2 * AF[i]; last block -> BF ----------------
// grid (NB/1024, 64 row-chunks of 64), block 256. Thread owns 4 columns (float4 loads,
// 512B per wave per row). AF chunk staged in LDS.
__global__ void cl_colsum(const float* __restrict__ k,
                          const float* __restrict__ bt,
                          const float* __restrict__ AF,
                          float* __restrict__ S,
                          float* __restrict__ BF,
                          int* __restrict__ counter,
                          int totalBlocks) {
  __shared__ float afs[64];
  __shared__ int lastFlag;

  const int j0 = blockIdx.x * 1024 + threadIdx.x * 4;
  const int r0 = blockIdx.y * 64;

  if (threadIdx.x < 64) afs[threadIdx.x] = AF[r0 + threadIdx.x];
  __syncthreads();

  float sx = 0.0f, sy = 0.0f, sz = 0.0f, sw = 0.0f;
  const float* kp = k + (size_t)r0 * NB + j0;
#pragma unroll 4
  for (int i = 0; i < 64; ++i) {
    float4 kv = *(const float4*)(kp + (size_t)i * NB);
    float a = afs[i];
    sx = fmaf(kv.x * kv.x, a, sx);
    sy = fmaf(kv.y * kv.y, a, sy);
    sz = fmaf(kv.z * kv.z, a, sz);
    sw = fmaf(kv.w * kv.w, a, sw);
  }
  atomicAdd(&S[j0 + 0], sx);
  atomicAdd(&S[j0 + 1], sy);
  atomicAdd(&S[j0 + 2], sz);
  atomicAdd(&S[j0 + 3], sw);
  __threadfence();
  __syncthreads();

  if (threadIdx.x == 0)
    lastFlag = (atomicAdd(counter, 1) == totalBlocks - 1) ? 1 : 0;
  __syncthreads();

  if (lastFlag) {
    __threadfence();
    for (int idx = threadIdx.x; idx < NB; idx += 256) {
      float sv  = ((volatile float*)S)[idx];
      float btv = bt[idx];
      BF[idx] = (btv * btv) / (1.0f + sv);
      S[idx]  = 0.0f;               // reset for next iteration
    }
    if (threadIdx.x == 0) *counter = 0;
  }
}

// ---------------- rowsum (WMMA): T[i] += sum_j k[i][j]^2 * BF[j]; last block -> AF ----------------
// grid (NA/128, 8 k-chunks of 512), block 256 = 8 waves; each wave owns 16 rows.
// b128 global loads feed TWO V_WMMA_F32_16X16X4_F32 each: wmma#1 takes column slots
// {kc,kc+1 | kc+4,kc+5}, wmma#2 {kc+2,kc+3 | kc+6,kc+7} (A and B use the same
// column->slot assignment, so the union covers kc..kc+7 exactly once).
__global__ void cl_rowsum(const float* __restrict__ k,
                          const float* __restrict__ AT,
                          const float* __restrict__ BF,
                          float* __restrict__ T,
                          float* __restrict__ AF,
                          int* __restrict__ counter,
                          int totalBlocks) {
  __shared__ float bfs[512];
  __shared__ int lastFlag;

  const int kBase = blockIdx.y * 512;
  for (int t = threadIdx.x; t < 512; t += 256)
    bfs[t] = BF[kBase + t];
  __syncthreads();

  const int wave = threadIdx.x >> 5;
  const int lane = threadIdx.x & 31;
  const int half = lane >> 4;       // 0: lanes 0-15 (K slots 0,1) ; 1: lanes 16-31 (K slots 2,3)
  const int lrow = lane & 15;
  const int rowBase = blockIdx.x * 128 + wave * 16;

  const float* rowPtr = k + (size_t)(rowBase + lrow) * NB + kBase;

  v8f c = {0.f, 0.f, 0.f, 0.f, 0.f, 0.f, 0.f, 0.f};

  for (int kc = 0; kc < 512; kc += 8) {
    const int co = kc + 4 * half;                       // 16B aligned
    float4 av = *(const float4*)(rowPtr + co);          // k[row][kBase+co .. +3]
    float4 bv = *(const float4*)&bfs[co];               // BF for the same 4 columns
    v2f a0; a0.x = av.x * av.x; a0.y = av.y * av.y;     // square on the fly (K = k*k)
    v2f a1; a1.x = av.z * av.z; a1.y = av.w * av.w;
    v2f b0; b0.x = bv.x; b0.y = bv.y;
    v2f b1; b1.x = bv.z; b1.y = bv.w;
    c = __builtin_amdgcn_wmma_f32_16x16x4_f32(
        false, a0, false, b0, (short)0, c, false, false);
    c = __builtin_amdgcn_wmma_f32_16x16x4_f32(
        false, a1, false, b1, (short)0, c, false, false);
  }

  // D columns are identical; lane 0 holds rows rowBase+0..7, lane 16 rows rowBase+8..15.
  if (lrow == 0) {
    float* dst = &T[rowBase + (half ? 8 : 0)];
#pragma unroll
    for (int m = 0; m < 8; ++m) atomicAdd(&dst[m], c[m]);
  }
  __threadfence();
  __syncthreads();

  if (threadIdx.x == 0)
    lastFlag = (atomicAdd(counter, 1) == totalBlocks - 1) ? 1 : 0;
  __syncthreads();

  if (lastFlag) {
    __threadfence();
    for (int idx = threadIdx.x; idx < NA; idx += 256) {
      float tv = ((volatile float*)T)[idx];
      AF[idx] = AT[idx] / (1.0f + tv);
      T[idx]  = 0.0f;               // reset for next iteration
    }
    if (threadIdx.x == 0) *counter = 0;
  }
}

// ---------------- final: C[i][j] = AF[i] * k[i][j]^2 * BF[j] ----------------
// one block per row; float4 streaming.
__global__ void cl_final(const float* __restrict__ k,
                         const float* __restrict__ AF,
                         const float* __restrict__ BF,
                         float* __restrict__ C) {
  const int row = blockIdx.x;
  const float af = AF[row];
  const float4* k4  = (const float4*)(k + (size_t)row * NB);
  const float4* bf4 = (const float4*)BF;
  float4* c4 = (float4*)(C + (size_t)row * NB);
  for (int t = threadIdx.x; t < NB / 4; t += 256) {
    float4 kv = k4[t];
    float4 bv = bf4[t];
    float4 o;
    o.x = af * kv.x * kv.x * bv.x;
    o.y = af * kv.y * kv.y * bv.y;
    o.z = af * kv.z * kv.z * bv.z;
    o.w = af * kv.w * kv.w * bv.w;
    c4[t] = o;
  }
}

extern "C" void kernel_launch(void* const* d_in, const int* in_sizes, int n_in,
                              void* d_out, int out_size, void* d_ws, size_t ws_size,
                              hipStream_t stream) {
  (void)in_sizes; (void)n_in; (void)out_size; (void)ws_size;
  const float* AT = (const float*)d_in[0];   // [4096]
  const float* k  = (const float*)d_in[1];   // [4096,4096]
  const float* bt = (const float*)d_in[2];   // [4096]
  float* out = (float*)d_out;

  float* ws = (float*)d_ws;
  float* S  = ws;              // [4096] colsum accumulator
  float* T  = ws + NA;         // [4096] rowsum accumulator
  float* AF = ws + 2 * NA;     // [4096]
  float* BF = ws + 3 * NA;     // [4096]
  int* counters = (int*)(ws + 4 * NA);

  cl_init<<<dim3(NA / 256), 256, 0, stream>>>(AT, S, T, AF, counters);

  for (int it = 0; it < N_ITERS; ++it) {
    cl_colsum<<<dim3(NB / 1024, 64), 256, 0, stream>>>(k, bt, AF, S, BF,
                                                       &counters[0], (NB / 1024) * 64);
    cl_rowsum<<<dim3(NA / 128, 8), 256, 0, stream>>>(k, AT, BF, T, AF,
                                                     &counters[1], (NA / 128) * 8);
  }

  cl_final<<<dim3(NA), 256, 0, stream>>>(k, AF, BF, out);
}